// KNN_89627377533638
// MI455X (gfx1250) — compile-verified
//
#include <hip/hip_runtime.h>
#include <math.h>

// ---------------------------------------------------------------------------
// KNN (K=16) via V_WMMA_F32_16X16X4_F32 distance tiles + per-lane sorted list.
// dist^2 = |q|^2 + |s|^2 - 2 q.s  ->  WMMA computes (|s|^2 - 2 q.s) with
//   A row  = (qx, qy, qz, 1)           (16x4, f32)
//   B col  = (-2sx, -2sy, -2sz, |s|^2) (4x16, f32)
// |q|^2 is a per-row constant -> irrelevant for ordering; added + sqrt at end.
// Support points staged into LDS with double-buffered async copies
// (GLOBAL_LOAD_ASYNC_TO_LDS_B128, ASYNCcnt).
// ---------------------------------------------------------------------------

typedef float v2f __attribute__((ext_vector_type(2)));
typedef float v8f __attribute__((ext_vector_type(8)));
typedef int   v4i_vs __attribute__((vector_size(4 * sizeof(int))));  // builtin ptr type

#define KNN_K    16
#define NQ       4096
#define NS       16384
#define NB       4
#define CHUNK    512                // support points staged per outer iteration
#define NCHUNK   (NS / CHUNK)
#define QPB      256                // queries per block (8 waves x 32 queries)
#define DT_PITCH 20                 // padded row pitch (floats) for D scratch

#if defined(__gfx1250__) && __has_builtin(__builtin_amdgcn_global_load_async_to_lds_b128)
#define USE_ASYNC_LDS 1
#else
#define USE_ASYNC_LDS 0
#endif

__device__ __forceinline__ void wait_async_zero()
{
#if defined(__gfx1250__)
#if __has_builtin(__builtin_amdgcn_s_wait_asynccnt)
    __builtin_amdgcn_s_wait_asynccnt(0);
#else
    __asm__ volatile("s_wait_asynccnt 0" ::: "memory");
#endif
#endif
}

// Stage one chunk of raw support xyz (CHUNK*12 bytes) into LDS.
__device__ __forceinline__ void stage_chunk(const float* __restrict__ sp,
                                            float* __restrict__ dstRaw,
                                            int cbase, int tid)
{
#if USE_ASYNC_LDS
    const char* g = (const char*)(sp + (size_t)cbase * 3);
    char*       l = (char*)dstRaw;
    // CHUNK*12B = 6144B = 384 x b128; 256 lanes do one, first 128 do a second.
    __builtin_amdgcn_global_load_async_to_lds_b128(
        (v4i_vs*)(g + tid * 16), (v4i_vs*)(l + tid * 16), 0, 0);
    if (tid < (CHUNK * 12 / 16) - 256) {
        __builtin_amdgcn_global_load_async_to_lds_b128(
            (v4i_vs*)(g + (256 + tid) * 16), (v4i_vs*)(l + (256 + tid) * 16), 0, 0);
    }
#else
#pragma unroll
    for (int k = 0; k < (CHUNK * 3) / 256; ++k)
        dstRaw[tid + k * 256] = sp[(size_t)cbase * 3 + tid + k * 256];
#endif
}

__global__ __launch_bounds__(256) void
knn_wmma_kernel(const float* __restrict__ query,
                const float* __restrict__ support,
                float* __restrict__ out_dist,
                int*   __restrict__ out_idx)
{
    __shared__ __align__(16) float s_raw[2][CHUNK * 3];            // raw xyz, dbl buf
    __shared__ __align__(16) float s_dt[8][2 * 32 * DT_PITCH];     // per-wave D scratch

    const int tid  = threadIdx.x;
    const int wave = tid >> 5;
    const int lane = tid & 31;
    const int hi   = lane >> 4;          // 0 = K0/K1 half, 1 = K2/K3 half
    const int col  = lane & 15;

    const int batch = blockIdx.x / (NQ / QPB);
    const int qblk  = (blockIdx.x % (NQ / QPB)) * QPB;

    const float* q  = query   + (size_t)batch * NQ * 3;
    const float* sp = support + (size_t)batch * NS * 3;

    // ---- loop-invariant A fragments (two 16-query tiles per wave) ----------
    // 32-bit A 16x4 layout: lanes 0-15 hold (K0,K1), lanes 16-31 hold (K2,K3).
    const int m0 = qblk + wave * 32 + col;
    const int m1 = m0 + 16;
    const float q0x = q[m0 * 3 + 0], q0y = q[m0 * 3 + 1], q0z = q[m0 * 3 + 2];
    const float q1x = q[m1 * 3 + 0], q1y = q[m1 * 3 + 1], q1z = q[m1 * 3 + 2];
    v2f a0, a1;
    a0.x = hi ? q0z : q0x;   a0.y = hi ? 1.0f : q0y;
    a1.x = hi ? q1z : q1x;   a1.y = hi ? 1.0f : q1y;

    // ---- this lane's own query (for top-k ownership) -----------------------
    const int   qown = qblk + wave * 32 + lane;
    const float ox = q[qown * 3 + 0], oy = q[qown * 3 + 1], oz = q[qown * 3 + 2];
    const float q2own = ox * ox + oy * oy + oz * oz;

    float best[KNN_K];
    int   bidx[KNN_K];
#pragma unroll
    for (int j = 0; j < KNN_K; ++j) { best[j] = 3.4e38f; bidx[j] = 0; }

    float* const dt0 = &s_dt[wave][0];
    float* const dt1 = dt0 + 32 * DT_PITCH;

    // ---- prologue: stage chunk 0 -------------------------------------------
    stage_chunk(sp, s_raw[0], 0, tid);
    wait_async_zero();
    __syncthreads();

    for (int c = 0; c < NCHUNK; ++c) {
        const int cur = c & 1;
        // kick off next chunk's async copy into the other buffer
        if (c + 1 < NCHUNK)
            stage_chunk(sp, s_raw[cur ^ 1], (c + 1) * CHUNK, tid);

        const float* raw = s_raw[cur];
        const int    cb  = c * CHUNK;

        for (int g = 0; g < CHUNK / 32; ++g) {
            // ---- two B fragments (32 supports), built on the fly -----------
            const int p0 = g * 32 + col;
            const int p1 = p0 + 16;
            const float x0 = raw[p0 * 3 + 0], y0 = raw[p0 * 3 + 1], z0 = raw[p0 * 3 + 2];
            const float x1 = raw[p1 * 3 + 0], y1 = raw[p1 * 3 + 1], z1 = raw[p1 * 3 + 2];
            v2f b0, b1;
            b0.x = hi ? (-2.0f * z0) : (-2.0f * x0);
            b0.y = hi ? (x0 * x0 + y0 * y0 + z0 * z0) : (-2.0f * y0);
            b1.x = hi ? (-2.0f * z1) : (-2.0f * x1);
            b1.y = hi ? (x1 * x1 + y1 * y1 + z1 * z1) : (-2.0f * y1);

            // ---- 4 independent WMMAs back-to-back --------------------------
            v8f cz = {};
            v8f d00 = __builtin_amdgcn_wmma_f32_16x16x4_f32(
                false, a0, false, b0, (short)0, cz, false, false);
            v8f d10 = __builtin_amdgcn_wmma_f32_16x16x4_f32(
                false, a1, false, b0, (short)0, cz, false, false);
            v8f d01 = __builtin_amdgcn_wmma_f32_16x16x4_f32(
                false, a0, false, b1, (short)0, cz, false, false);
            v8f d11 = __builtin_amdgcn_wmma_f32_16x16x4_f32(
                false, a1, false, b1, (short)0, cz, false, false);

            // ---- scatter D tiles into per-query LDS rows -------------------
            // element (vgpr r, lane) = (row r + 8*hi, col lane&15)
#pragma unroll
            for (int r = 0; r < 8; ++r) {
                dt0[(r + 8 * hi) * DT_PITCH + col]      = d00[r];
                dt0[(16 + r + 8 * hi) * DT_PITCH + col] = d10[r];
                dt1[(r + 8 * hi) * DT_PITCH + col]      = d01[r];
                dt1[(16 + r + 8 * hi) * DT_PITCH + col] = d11[r];
            }
            __builtin_amdgcn_wave_barrier();
            __asm__ volatile("s_wait_dscnt 0" ::: "memory");   // wave-local hazard

            // ---- each lane scans its own query's 32 candidates -------------
            // (fully inline: no call boundary, keeps best[]/bidx[] in VGPRs)
#pragma unroll
            for (int t = 0; t < 2; ++t) {
                const float* row   = (t == 0 ? dt0 : dt1) + lane * DT_PITCH;
                const int    ibase = cb + g * 32 + t * 16;
#pragma unroll
                for (int n4 = 0; n4 < 4; ++n4) {
                    const float4 v4 = *(const float4*)(row + n4 * 4);
                    const float vv[4] = { v4.x, v4.y, v4.z, v4.w };
#pragma unroll
                    for (int e = 0; e < 4; ++e) {
                        const float v = vv[e];
                        if (v < best[KNN_K - 1]) {              // guarded insert
                            best[KNN_K - 1] = v;
                            bidx[KNN_K - 1] = ibase + n4 * 4 + e;
#pragma unroll
                            for (int j = KNN_K - 1; j > 0; --j) {
                                const bool  sw = best[j] < best[j - 1];
                                const float tv = best[j - 1];
                                const int   ti = bidx[j - 1];
                                best[j - 1] = sw ? best[j] : best[j - 1];
                                bidx[j - 1] = sw ? bidx[j] : bidx[j - 1];
                                best[j]     = sw ? tv : best[j];
                                bidx[j]     = sw ? ti : bidx[j];
                            }
                        }
                    }
                }
            }
            // DS ops within a wave are in-order: next group's stores cannot
            // pass these loads, so no extra wait needed here.
        }

        // own async copy done, then block-wide visibility for next iteration
        wait_async_zero();
        __syncthreads();
    }

    // ---- finalize: add |q|^2, sqrt, emit (sorted ascending) ----------------
#pragma unroll
    for (int j = 0; j < KNN_K; ++j) {
        float d2 = best[j] + q2own;
        d2 = d2 > 0.0f ? d2 : 0.0f;
        const size_t o = (size_t)(batch * NQ + qown) * KNN_K + j;
        out_dist[o] = sqrtf(d2);
        out_idx[o]  = bidx[j];
    }
}

extern "C" void kernel_launch(void* const* d_in, const int* in_sizes, int n_in,
                              void* d_out, int out_size, void* d_ws, size_t ws_size,
                              hipStream_t stream)
{
    (void)in_sizes; (void)n_in; (void)out_size; (void)d_ws; (void)ws_size;

    const float* query   = (const float*)d_in[0];   // [4, 4096, 3] f32
    const float* support = (const float*)d_in[1];   // [4, 16384, 3] f32

    float* out_dist = (float*)d_out;                          // [4,4096,16] f32
    int*   out_idx  = (int*)d_out + (size_t)NB * NQ * KNN_K;  // [4,4096,16] i32

    dim3 grid(NB * (NQ / QPB));   // 64 blocks
    dim3 block(256);              // 8 wave32 per block
    knn_wmma_kernel<<<grid, block, 0, stream>>>(query, support, out_dist, out_idx);
}